// NodeMeanAggregator_73469710565689
// MI455X (gfx1250) — compile-verified
//
#include <hip/hip_runtime.h>
#include <stdint.h>

// NodeMeanAggregator: out[n, :] = mean_k table[idx[n,k], :]
//   table: [500000, 128] fp32 (row = 512 B), idx: [100000, 32] int32.
// Memory-bound gather; one wave32 per node (DEGREE == 32 == wave size).
// CDNA5 path: GLOBAL_LOAD_ASYNC_TO_LDS_B128 double-buffered via ASYNCcnt.

#define DEGREE          32
#define EMBED_DIM       128
#define ROW_BYTES       (EMBED_DIM * 4)     // 512
#define WAVES_PER_BLOCK 8
#define CHUNK           8                   // rows per async chunk
#define NCHUNK          (DEGREE / CHUNK)    // 4

__device__ __forceinline__ void wait_asynccnt_le8() {
    asm volatile("s_wait_asynccnt 0x8" ::: "memory");
}
__device__ __forceinline__ void wait_asynccnt_0() {
    asm volatile("s_wait_asynccnt 0x0" ::: "memory");
}

__global__ __launch_bounds__(WAVES_PER_BLOCK * 32)
void NodeMeanAggregator_kernel(const float* __restrict__ table,
                               const int*   __restrict__ edges,
                               float*       __restrict__ out,
                               int n_nodes)
{
    // 8 waves * 2 buffers * 8 rows * 512 B = 64 KB LDS per block.
    __shared__ float4 smem[WAVES_PER_BLOCK * 2 * CHUNK * 32];

    const int lane = threadIdx.x & 31;
    const int wave = threadIdx.x >> 5;
    const int node = blockIdx.x * WAVES_PER_BLOCK + wave;
    if (node >= n_nodes) return;           // wave-uniform branch

    // Lane l holds this node's l-th hyperedge index (DEGREE == 32).
    const int idx_l = edges[node * DEGREE + lane];

    float4* buf0 = &smem[(wave * 2 + 0) * (CHUNK * 32)];
    float4* buf1 = &smem[(wave * 2 + 1) * (CHUNK * 32)];
    // Flat LDS address: low 32 bits of the generic pointer are the LDS byte
    // offset (ISA 10.2 aperture layout). Each lane targets its 16 B slice.
    const uint32_t lds0 = (uint32_t)(uintptr_t)buf0 + (uint32_t)(lane << 4);
    const uint32_t lds1 = (uint32_t)(uintptr_t)buf1 + (uint32_t)(lane << 4);

    // Issue 8 async row copies: row k of chunk c -> ldsbase + k*512.
    // Per-lane global offset = row*512 + lane*16 (< 2^31, GVS i32 voffset ok).
    auto issue_chunk = [&](int c, uint32_t ldsbase) {
#pragma unroll
        for (int k = 0; k < CHUNK; ++k) {
            const int row = __builtin_amdgcn_readlane(idx_l, c * CHUNK + k);
            const uint32_t goff = ((uint32_t)row << 9) | ((uint32_t)lane << 4);
            asm volatile("global_load_async_to_lds_b128 %0, %1, %2"
                         :
                         : "v"(ldsbase + (uint32_t)(k * ROW_BYTES)),
                           "v"(goff),
                           "s"(table)
                         : "memory");
        }
    };

    float4 acc = make_float4(0.f, 0.f, 0.f, 0.f);
    auto consume_chunk = [&](const float4* buf) {
#pragma unroll
        for (int k = 0; k < CHUNK; ++k) {
            const float4 v = buf[k * 32 + lane];   // ds_load_b128
            acc.x += v.x; acc.y += v.y; acc.z += v.z; acc.w += v.w;
        }
    };

    // Software pipeline: 2 chunks (16 rows, 8 KB) in flight at all times.
    issue_chunk(0, lds0);
    issue_chunk(1, lds1);

    wait_asynccnt_le8();                   // chunk 0 complete (in-order)
    consume_chunk(buf0);
    issue_chunk(2, lds0);

    wait_asynccnt_le8();                   // chunk 1 complete
    consume_chunk(buf1);
    issue_chunk(3, lds1);

    wait_asynccnt_le8();                   // chunk 2 complete
    consume_chunk(buf0);

    wait_asynccnt_0();                     // chunk 3 complete
    consume_chunk(buf1);

    const float s = 1.0f / (float)DEGREE;
    const float4 r = make_float4(acc.x * s, acc.y * s, acc.z * s, acc.w * s);
    // Coalesced 512 B store per wave.
    reinterpret_cast<float4*>(out)[node * 32 + lane] = r;
}

extern "C" void kernel_launch(void* const* d_in, const int* in_sizes, int n_in,
                              void* d_out, int out_size, void* d_ws, size_t ws_size,
                              hipStream_t stream) {
    const float* table = (const float*)d_in[0];   // [500000, 128] fp32
    const int*   edges = (const int*)d_in[1];     // [100000, 32] int32
    float*       out   = (float*)d_out;           // [100000, 128] fp32

    const int n_nodes = in_sizes[1] / DEGREE;
    const int blocks  = (n_nodes + WAVES_PER_BLOCK - 1) / WAVES_PER_BLOCK;

    NodeMeanAggregator_kernel<<<blocks, WAVES_PER_BLOCK * 32, 0, stream>>>(
        table, edges, out, n_nodes);
}